// RelationKnowledgeDistillation_7275674599738
// MI455X (gfx1250) — compile-verified
//
#include <hip/hip_runtime.h>
#include <hip/hip_bf16.h>

// ---------------------------------------------------------------------------
// RKD (relational knowledge distillation) loss on MI455X (gfx1250, wave32).
//
// Math reduction: both the pairwise-distance loss and the angle loss are
// functions of the Gram matrices Gs = Es*Es^T, Gt = Et*Et^T (256x256):
//   dist^2(i,j) = G_ii + G_jj - 2*G_ij + 2*eps*(s_i - s_j) + D*eps^2
//   dots[j,a,b] = G_ab - G_aj - G_jb + G_jj
//   norm[j,a]^2 = G_aa + G_jj - 2*G_aj
// The Gram GEMM (256x512x256, fp32) is done on the WMMA pipe with
// V_WMMA_F32_16X16X4_F32 (fp32 A/B, fp32 accumulate -> no precision loss).
// The O(N^3/6) angle triplet loop is VALU work (fma + acos), parallelized
// one block per vertex j with row-j data staged in LDS.
// ---------------------------------------------------------------------------

typedef float v2f __attribute__((ext_vector_type(2)));
typedef float v8f __attribute__((ext_vector_type(8)));

namespace {
constexpr int   kN       = 256;
constexpr int   kD       = 512;
constexpr float kEpsPd   = 1e-6f;   // added to the difference vector
constexpr float kInvEpsC = 1e8f;    // 1 / EPS_COS (denominator floor)
constexpr float kClamp   = 1e-7f;   // acos clamp margin
constexpr float kDistW   = 1.0f;
constexpr float kAngleW  = 2.0f;
constexpr float kTriplets = 2763520.0f;  // 256*255*254/6
}

// --- row sums (for the +eps term of pairwise_distance) ----------------------
// 512 waves total (256 rows x 2 matrices), one wave per row.
__global__ void rkd_row_sums(const float* __restrict__ Es,
                             const float* __restrict__ Et,
                             float* __restrict__ ss, float* __restrict__ st) {
  int wave = blockIdx.x * (blockDim.x >> 5) + (threadIdx.x >> 5);
  int lane = threadIdx.x & 31;
  const float* E = (wave < kN) ? Es : Et;
  float*       s = (wave < kN) ? ss : st;
  int row = wave & (kN - 1);
  float v = 0.0f;
  for (int c = lane; c < kD; c += 32) v += E[row * kD + c];
  for (int off = 16; off > 0; off >>= 1) v += __shfl_xor(v, off, 32);
  if (lane == 0) s[row] = v;
}

// --- Gram matrices via V_WMMA_F32_16X16X4_F32 -------------------------------
// One wave per 16x16 output tile; 2 matrices * 16*16 tiles = 512 waves.
// A (16x4, M x K) fp32 layout: lanes 0-15 row M=lane hold K={0,1} in the two
// VGPRs, lanes 16-31 hold K={2,3}; B (4x16, K x N) mirrors this with N across
// lanes. So each lane fetches its two K values as one aligned float2.
__global__ void rkd_gram_wmma(const float* __restrict__ Es,
                              const float* __restrict__ Et,
                              float* __restrict__ Gs, float* __restrict__ Gt) {
  int wave = blockIdx.x * (blockDim.x >> 5) + (threadIdx.x >> 5);
  int lane = threadIdx.x & 31;
  const float* E = (wave < kN) ? Es : Et;
  float*       G = (wave < kN) ? Gs : Gt;
  int tile = wave & 255;
  int mt = (tile >> 4) << 4;          // row-tile base
  int nt = (tile & 15) << 4;          // col-tile base
  int l15  = lane & 15;
  int koff = (lane < 16) ? 0 : 2;     // which K pair this lane holds
  const float* arow = E + (mt + l15) * kD + koff;
  const float* brow = E + (nt + l15) * kD + koff;

  v8f acc = {};
  #pragma unroll 4
  for (int k = 0; k < kD; k += 4) {
    v2f a = *(const v2f*)(arow + k);
    v2f b = *(const v2f*)(brow + k);
    // D = A(16x4) * B(4x16) + C ; fp32 in, fp32 accumulate
    acc = __builtin_amdgcn_wmma_f32_16x16x4_f32(
        /*neg_a=*/false, a, /*neg_b=*/false, b,
        /*c_mod=*/(short)0, acc, /*reuse_a=*/false, /*reuse_b=*/false);
  }

  // C/D layout: VGPR g -> row M=g (lanes 0-15) / M=g+8 (lanes 16-31)
  int rhalf = (lane < 16) ? 0 : 8;
  #pragma unroll
  for (int g = 0; g < 8; ++g) {
    G[(mt + g + rhalf) * kN + nt + l15] = acc[g];
  }
}

// --- reciprocal norms r[j,a] = 1/||e_a - e_j|| ------------------------------
// min(r_ja*r_jk, 1e8) later reproduces max(n_ja*n_jk, 1e-8) exactly.
__global__ void rkd_recip_norms(const float* __restrict__ Gs,
                                const float* __restrict__ Gt,
                                float* __restrict__ Rs, float* __restrict__ Rt) {
  int b = blockIdx.x;
  const float* G = (b < kN) ? Gs : Gt;
  float*       R = (b < kN) ? Rs : Rt;
  int j = b & (kN - 1);
  int a = threadIdx.x;
  float v = G[a * kN + a] + G[j * kN + j] - 2.0f * G[j * kN + a];
  v = fmaxf(v, 0.0f);                  // guard rounding; v==0 -> r=inf -> min-clamped
  R[j * kN + a] = 1.0f / sqrtf(v);
}

__device__ __forceinline__ float smooth_l1(float x) {
  float ax = fabsf(x);
  return (ax < 1.0f) ? 0.5f * ax * ax : ax - 0.5f;
}

// --- distance loss partials: block j sums over i<j --------------------------
__global__ void rkd_dist_partial(const float* __restrict__ Gs,
                                 const float* __restrict__ Gt,
                                 const float* __restrict__ ss,
                                 const float* __restrict__ st,
                                 float* __restrict__ pd) {
  __shared__ float red[kN];
  int j = blockIdx.x, i = threadIdx.x;
  float f = 0.0f;
  if (i < j) {
    const float cterm = (float)kD * kEpsPd * kEpsPd;
    float ds2 = Gs[i * kN + i] + Gs[j * kN + j] - 2.0f * Gs[i * kN + j]
              + 2.0f * kEpsPd * (ss[i] - ss[j]) + cterm;
    float dt2 = Gt[i * kN + i] + Gt[j * kN + j] - 2.0f * Gt[i * kN + j]
              + 2.0f * kEpsPd * (st[i] - st[j]) + cterm;
    f = smooth_l1(sqrtf(fmaxf(ds2, 0.0f)) - sqrtf(fmaxf(dt2, 0.0f)));
  }
  red[i] = f;
  __syncthreads();
  for (int s = kN >> 1; s > 0; s >>= 1) {
    if (i < s) red[i] += red[i + s];
    __syncthreads();
  }
  if (i == 0) pd[j] = red[0];
}

// --- angle loss partials: block j iterates triples (i < j < k) --------------
__global__ void rkd_angle_partial(const float* __restrict__ Gs,
                                  const float* __restrict__ Gt,
                                  const float* __restrict__ Rs,
                                  const float* __restrict__ Rt,
                                  float* __restrict__ pa) {
  __shared__ float gsj[kN], gtj[kN], rsj[kN], rtj[kN];
  __shared__ float red[kN];
  int j = blockIdx.x, t = threadIdx.x;
  gsj[t] = Gs[j * kN + t];
  gtj[t] = Gt[j * kN + t];
  rsj[t] = Rs[j * kN + t];
  rtj[t] = Rt[j * kN + t];
  __syncthreads();

  float gsjj = gsj[j], gtjj = gtj[j];
  int W = kN - 1 - j;                  // number of valid k values (k in (j, N))
  int total = j * W;                   // pairs (i,k) with i<j<k ; max 16256
  float acc = 0.0f;
  for (int p = t; p < total; p += blockDim.x) {
    int i = p / W;
    int k = j + 1 + (p - i * W);

    float dots = Gs[i * kN + k] - gsj[i] - gsj[k] + gsjj;
    float cs = dots * fminf(rsj[i] * rsj[k], kInvEpsC);
    cs = fminf(fmaxf(cs, -1.0f + kClamp), 1.0f - kClamp);
    float as_ = acosf(cs);

    float dott = Gt[i * kN + k] - gtj[i] - gtj[k] + gtjj;
    float ct = dott * fminf(rtj[i] * rtj[k], kInvEpsC);
    ct = fminf(fmaxf(ct, -1.0f + kClamp), 1.0f - kClamp);
    float at_ = acosf(ct);

    acc += smooth_l1(as_ - at_);
  }
  red[t] = acc;
  __syncthreads();
  for (int s = kN >> 1; s > 0; s >>= 1) {
    if (t < s) red[t] += red[t + s];
    __syncthreads();
  }
  if (t == 0) pa[j] = red[0];
}

// --- deterministic final reduction + loss combine ---------------------------
__global__ void rkd_finalize(const float* __restrict__ pd,
                             const float* __restrict__ pa,
                             float* __restrict__ out) {
  __shared__ float rd[kN], ra[kN];
  int t = threadIdx.x;
  rd[t] = pd[t];
  ra[t] = pa[t];
  __syncthreads();
  for (int s = kN >> 1; s > 0; s >>= 1) {
    if (t < s) { rd[t] += rd[t + s]; ra[t] += ra[t + s]; }
    __syncthreads();
  }
  if (t == 0) {
    float dist_loss  = 2.0f * rd[0] / (float)(kN * kN);  // mirrored upper tri / N^2
    float angle_loss = ra[0] / kTriplets;
    out[0] = kDistW * dist_loss + kAngleW * angle_loss;
    out[1] = dist_loss;
    out[2] = angle_loss;
  }
}

extern "C" void kernel_launch(void* const* d_in, const int* in_sizes, int n_in,
                              void* d_out, int out_size, void* d_ws, size_t ws_size,
                              hipStream_t stream) {
  const float* Es = (const float*)d_in[0];
  const float* Et = (const float*)d_in[1];
  float* out = (float*)d_out;

  // Workspace layout (floats): ~1.03 MB total
  float* ws = (float*)d_ws;
  float* Gs = ws;                 // 65536
  float* Gt = ws + 65536;         // 65536
  float* Rs = ws + 131072;        // 65536
  float* Rt = ws + 196608;        // 65536
  float* ss = ws + 262144;        // 256
  float* st = ws + 262400;        // 256
  float* pd = ws + 262656;        // 256
  float* pa = ws + 262912;        // 256

  rkd_row_sums<<<64, 256, 0, stream>>>(Es, Et, ss, st);          // 512 waves
  rkd_gram_wmma<<<64, 256, 0, stream>>>(Es, Et, Gs, Gt);         // 512 tile-waves
  rkd_recip_norms<<<512, 256, 0, stream>>>(Gs, Gt, Rs, Rt);
  rkd_dist_partial<<<256, 256, 0, stream>>>(Gs, Gt, ss, st, pd);
  rkd_angle_partial<<<256, 256, 0, stream>>>(Gs, Gt, Rs, Rt, pa);
  rkd_finalize<<<1, 256, 0, stream>>>(pd, pa, out);
}